// NTM_write_head_85074712199499
// MI455X (gfx1250) — compile-verified
//
#include <hip/hip_runtime.h>
#include <math.h>

// ---------------------------------------------------------------------------
// NTM write head, MI455X (gfx1250, wave32).
// B=1024, N=1024, M=64, C=512, L=198.  Bandwidth-bound (~0.8 GB @ 23.3 TB/s).
// GEMM (emb @ W_lin^T + b) uses V_WMMA_F32_16X16X4_F32, one wave per 16x16 tile.
// ---------------------------------------------------------------------------

#define Bn   1024
#define Nn   1024
#define Mn   64
#define Cn   512
#define Ln   198
#define LPAD 208      // 13 * 16
#define EPSF 1e-8f

typedef float v2f __attribute__((ext_vector_type(2)));
typedef float v8f __attribute__((ext_vector_type(8)));

// ----------------------------- helpers -------------------------------------

__device__ __forceinline__ float softplus_f(float x) {
    // numerically stable log(1+exp(x))
    return (x > 0.f) ? x + log1pf(expf(-x)) : log1pf(expf(x));
}
__device__ __forceinline__ float sigmoid_f(float x) {
    return 1.f / (1.f + expf(-x));
}
__device__ __forceinline__ float wave_reduce_add(float v) {
    #pragma unroll
    for (int off = 16; off > 0; off >>= 1) v += __shfl_xor(v, off, 32);
    return v;   // butterfly: all lanes hold the sum
}
__device__ __forceinline__ float wave_reduce_max(float v) {
    #pragma unroll
    for (int off = 16; off > 0; off >>= 1) v = fmaxf(v, __shfl_xor(v, off, 32));
    return v;
}
// blockDim.x == 1024 (32 waves) assumed.
__device__ __forceinline__ float block_reduce_add(float v, float* red) {
    int lane = threadIdx.x & 31, w = threadIdx.x >> 5;
    v = wave_reduce_add(v);
    if (lane == 0) red[w] = v;
    __syncthreads();
    if (w == 0) {
        float r = red[lane];
        r = wave_reduce_add(r);
        if (lane == 0) red[0] = r;
    }
    __syncthreads();
    float out = red[0];
    __syncthreads();
    return out;
}
__device__ __forceinline__ float block_reduce_max(float v, float* red) {
    int lane = threadIdx.x & 31, w = threadIdx.x >> 5;
    v = wave_reduce_max(v);
    if (lane == 0) red[w] = v;
    __syncthreads();
    if (w == 0) {
        float r = red[lane];
        r = wave_reduce_max(r);
        if (lane == 0) red[0] = r;
    }
    __syncthreads();
    float out = red[0];
    __syncthreads();
    return out;
}

// ---------------------------------------------------------------------------
// Kernel 1: preact[B][LPAD] = emb[B][C] @ Wlin[L][C]^T + b  via WMMA f32 16x16x4.
// One wave per 16x16 output tile; tile grid = (B/16) x ceil(L/16) = 64 x 13 = 832.
// f32 A(16x4): lane<16 holds K={0,1}, lane>=16 holds K={2,3}; M = lane&15.
// f32 B(4x16): same lane->K mapping; N = lane&15.
// D(16x16):    N = lane&15; M = vgpr + 8*(lane>>4).
// ---------------------------------------------------------------------------
__global__ void gemm_wmma_kernel(const float* __restrict__ emb,
                                 const float* __restrict__ Wlin,
                                 const float* __restrict__ blin,
                                 float* __restrict__ preact) {
    const int tilesN = LPAD / 16;                       // 13
    int tile = blockIdx.x * (blockDim.x >> 5) + (threadIdx.x >> 5);
    if (tile >= (Bn / 16) * tilesN) return;             // whole-wave uniform
    int tm = tile / tilesN;                             // batch tile
    int tn = tile % tilesN;                             // L tile

    int lane = threadIdx.x & 31;
    int half = lane >> 4;                               // 0 or 1
    int l16  = lane & 15;

    int row    = tm * 16 + l16;                         // A row (batch)
    int colRaw = tn * 16 + l16;                         // B row (output col)
    int col    = (colRaw < Ln) ? colRaw : (Ln - 1);     // clamp loads for pad cols

    const float* Arow = emb  + (size_t)row * Cn;
    const float* Brow = Wlin + (size_t)col * Cn;

    v8f acc = {};
    #pragma unroll 4
    for (int k0 = 0; k0 < Cn; k0 += 4) {
        int k = k0 + half * 2;
        v2f a, b;
        a.x = Arow[k];  a.y = Arow[k + 1];
        b.x = Brow[k];  b.y = Brow[k + 1];
        acc = __builtin_amdgcn_wmma_f32_16x16x4_f32(
            /*neg_a=*/false, a, /*neg_b=*/false, b,
            /*c_mod=*/(short)0, acc, /*reuse_a=*/false, /*reuse_b=*/false);
    }

    float bias = blin[col];
    #pragma unroll
    for (int v = 0; v < 8; ++v) {
        int m = tm * 16 + v + 8 * half;
        preact[(size_t)m * LPAD + colRaw] = acc[v] + bias;  // pad cols hold junk, never read
    }
}

// ---------------------------------------------------------------------------
// Kernel 2: activations, one wave per batch. Folds beta into normalized key.
// ---------------------------------------------------------------------------
__global__ void act_kernel(const float* __restrict__ preact,
                           float* __restrict__ kns, float* __restrict__ e,
                           float* __restrict__ a,   float* __restrict__ g,
                           float* __restrict__ gamma, float* __restrict__ s) {
    int b    = (blockIdx.x * blockDim.x + threadIdx.x) >> 5;
    int lane = threadIdx.x & 31;
    if (b >= Bn) return;
    const float* o = preact + (size_t)b * LPAD;

    float k0 = o[lane], k1 = o[lane + 32];
    float ss = wave_reduce_add(k0 * k0 + k1 * k1);

    float beta  = softplus_f(o[Mn]);                    // o[64]
    float scale = beta / (sqrtf(ss) + EPSF);
    kns[b * Mn + lane]      = k0 * scale;
    kns[b * Mn + lane + 32] = k1 * scale;

    e[b * Mn + lane]      = sigmoid_f(o[Mn + 6 + lane]);
    e[b * Mn + lane + 32] = sigmoid_f(o[Mn + 6 + lane + 32]);
    a[b * Mn + lane]      = o[2 * Mn + 6 + lane];
    a[b * Mn + lane + 32] = o[2 * Mn + 6 + lane + 32];

    if (lane == 0) {
        g[b]     = sigmoid_f(o[Mn + 1]);
        gamma[b] = 1.f + softplus_f(o[Mn + 5]);
        float o0 = o[Mn + 2], o1 = o[Mn + 3], o2 = o[Mn + 4];
        float mx = fmaxf(o0, fmaxf(o1, o2));
        float e0 = expf(o0 - mx), e1 = expf(o1 - mx), e2 = expf(o2 - mx);
        float inv = 1.f / (e0 + e1 + e2);
        s[b * 3 + 0] = e0 * inv;
        s[b * 3 + 1] = e1 * inv;
        s[b * 3 + 2] = e2 * inv;
    }
}

// ---------------------------------------------------------------------------
// Kernel 3: score[b][n] = beta * cos = <mem_row, kns_b> / (||mem_row|| + eps).
// One wave per row; float2 loads; butterfly reductions. 268 MB streaming read.
// ---------------------------------------------------------------------------
__global__ void score_kernel(const float* __restrict__ mem,
                             const float* __restrict__ kns,
                             float* __restrict__ score) {
    size_t rowId = ((size_t)blockIdx.x * blockDim.x + threadIdx.x) >> 5; // b*N + n
    int lane = threadIdx.x & 31;
    int b = (int)(rowId >> 10);

    const float2* row = (const float2*)(mem + rowId * Mn);
    const float2* kv  = (const float2*)(kns + (size_t)b * Mn);
    float2 m = row[lane];
    float2 k = kv[lane];

    float dot = m.x * k.x + m.y * k.y;
    float ss  = m.x * m.x + m.y * m.y;
    #pragma unroll
    for (int off = 16; off > 0; off >>= 1) {
        dot += __shfl_xor(dot, off, 32);
        ss  += __shfl_xor(ss,  off, 32);
    }
    if (lane == 0) score[rowId] = dot / (sqrtf(ss) + EPSF);
}

// ---------------------------------------------------------------------------
// Kernel 4: per-batch addressing; one 1024-thread block per batch, in-place on W.
// ---------------------------------------------------------------------------
__global__ void addressing_kernel(const float* __restrict__ Wold,
                                  const float* __restrict__ g_arr,
                                  const float* __restrict__ s_arr,
                                  const float* __restrict__ gamma_arr,
                                  float* __restrict__ W) {
    __shared__ float wg[Nn];
    __shared__ float red[32];
    int b = blockIdx.x, n = threadIdx.x;
    size_t idx = (size_t)b * Nn + n;

    float x  = W[idx];                                  // beta*cos (scratch)
    float mx = block_reduce_max(x, red);
    float ex = expf(x - mx);
    float S  = block_reduce_add(ex, red);

    float g   = g_arr[b];
    float wgv = g * (ex / S) + (1.f - g) * Wold[idx];
    wg[n] = wgv;
    __syncthreads();

    float s0 = s_arr[b * 3 + 0], s1 = s_arr[b * 3 + 1], s2 = s_arr[b * 3 + 2];
    float ws = s0 * wg[(n + Nn - 1) & (Nn - 1)] + s1 * wgv + s2 * wg[(n + 1) & (Nn - 1)];

    float wp = powf(ws + EPSF, gamma_arr[b]);
    float P  = block_reduce_add(wp, red);
    W[idx] = wp / P;
}

// ---------------------------------------------------------------------------
// Kernel 5: new_memory = memory*(1 - W⊗e) + W⊗a.  float4 streaming, 536 MB.
// ---------------------------------------------------------------------------
__global__ void update_kernel(const float* __restrict__ mem,
                              const float* __restrict__ W,
                              const float* __restrict__ e,
                              const float* __restrict__ a,
                              float* __restrict__ newmem) {
    size_t idx = (size_t)blockIdx.x * blockDim.x + threadIdx.x;   // float4 index
    int    m4  = (int)(idx & 15);                                 // 16 float4 / row
    size_t row = idx >> 4;                                        // b*N + n
    int    b   = (int)(row >> 10);

    float  w  = W[row];
    float4 mv = ((const float4*)mem)[idx];
    float4 ev = ((const float4*)(e + (size_t)b * Mn))[m4];
    float4 av = ((const float4*)(a + (size_t)b * Mn))[m4];

    float4 r;
    r.x = mv.x * (1.f - w * ev.x) + w * av.x;
    r.y = mv.y * (1.f - w * ev.y) + w * av.y;
    r.z = mv.z * (1.f - w * ev.z) + w * av.z;
    r.w = mv.w * (1.f - w * ev.w) + w * av.w;
    ((float4*)newmem)[idx] = r;
}

// ---------------------------------------------------------------------------
extern "C" void kernel_launch(void* const* d_in, const int* in_sizes, int n_in,
                              void* d_out, int out_size, void* d_ws, size_t ws_size,
                              hipStream_t stream) {
    const float* emb  = (const float*)d_in[0];   // [B,C]
    const float* Wold = (const float*)d_in[1];   // [B,N]
    const float* mem  = (const float*)d_in[2];   // [B,N,M]
    const float* Wlin = (const float*)d_in[3];   // [L,C]
    const float* blin = (const float*)d_in[4];   // [L]

    float* W      = (float*)d_out;                         // [B,N]
    float* newmem = (float*)d_out + (size_t)Bn * Nn;       // [B,N,M]

    float* ws     = (float*)d_ws;
    float* preact = ws;                                    // B*LPAD  = 212992
    float* kns    = preact + (size_t)Bn * LPAD;            // B*M     =  65536
    float* e      = kns    + (size_t)Bn * Mn;              // B*M
    float* a      = e      + (size_t)Bn * Mn;              // B*M
    float* g      = a      + (size_t)Bn * Mn;              // B
    float* gamma  = g      + Bn;                           // B
    float* s      = gamma  + Bn;                           // B*3

    // 1. GEMM: 832 tiles, 8 waves/block -> 104 blocks of 256 threads.
    gemm_wmma_kernel<<<104, 256, 0, stream>>>(emb, Wlin, blin, preact);

    // 2. Activations: 1 wave/batch, 8 waves/block -> 128 blocks.
    act_kernel<<<128, 256, 0, stream>>>(preact, kns, e, a, g, gamma, s);

    // 3. Scores: B*N waves, 8/block -> 131072 blocks. Writes into W as scratch.
    score_kernel<<<(Bn * Nn) / 8, 256, 0, stream>>>(mem, kns, W);

    // 4. Addressing: one 1024-thread block per batch, in-place on W.
    addressing_kernel<<<Bn, Nn, 0, stream>>>(Wold, g, s, gamma, W);

    // 5. Memory update: B*N*M/4 float4s, 256 threads/block -> 65536 blocks.
    update_kernel<<<(Bn * Nn * Mn / 4) / 256, 256, 0, stream>>>(mem, W, e, a, newmem);
}